// RNNModel_64398739636719
// MI455X (gfx1250) — compile-verified
//
#include <hip/hip_runtime.h>

// GRU-like recurrence, last layer only (reference overwrites X per layer).
// H=2048, T=128 steps, layer 3 of 4. All matvecs use V_WMMA_F32_16X16X4_F32
// (fp32 in/out, full precision); B = input-vector chunk broadcast across all
// 16 N columns so every column of D holds the same dot product.
// Software-pipelined: chunk c+1 operands are loaded while chunk c's 6 WMMAs
// issue, so loads and WMMAs from adjacent iterations overlap.

#define H     2048
#define T     128
#define LAYER 3

typedef __attribute__((ext_vector_type(2))) float v2f;
typedef __attribute__((ext_vector_type(8))) float v8f;

__device__ __forceinline__ v8f wmma_f32(v2f a, v2f b, v8f c) {
    // 8 args: (neg_a, A, neg_b, B, c_mod, C, reuse_a, reuse_b)
    return __builtin_amdgcn_wmma_f32_16x16x4_f32(
        false, a, false, b, (short)0, c, false, false);
}

__global__ void rnn_init_kernel(const float* __restrict__ x,
                                float* __restrict__ out,
                                float* __restrict__ hzero) {
    int i = blockIdx.x * blockDim.x + threadIdx.x;
    if (i < H) {
        out[i]   = x[i];   // X[0] = original input
        hzero[i] = 0.0f;   // h starts at zeros
    }
}

__global__ __launch_bounds__(256)
void rnn_step_kernel(const float* __restrict__ wxh,  // [2H,H] layer base
                     const float* __restrict__ whh,  // [2H,H]
                     const float* __restrict__ wxr,  // [H,H]
                     const float* __restrict__ whr,  // [H,H]
                     const float* __restrict__ bxh,  // [2H]
                     const float* __restrict__ bhh,  // [2H]
                     const float* __restrict__ bxr,  // [H]
                     const float* __restrict__ bhr,  // [H]
                     const float* __restrict__ xt,   // [H]
                     const float* __restrict__ h,    // [H] (zeros at t=0)
                     float* __restrict__ out)        // [H] = X[t+1]
{
    __shared__ float red[8][4][16];  // [wave][{z,r,e,f}][row]

    const int tid  = threadIdx.x;
    const int lane = tid & 31;
    const int wave = tid >> 5;          // 8 waves, K segment of 256 each
    const int ml   = lane & 15;         // A-matrix row within 16-tile
    const int koff = (lane >> 4) << 1;  // lanes 16-31 hold K = {2,3}
    const int row0 = blockIdx.x * 16;   // 128 blocks cover H rows

    const int kb = wave * 256 + koff;

    const float* pZx = wxh + (size_t)(row0 + ml) * H + kb;       // z-rows of Wxh
    const float* pZh = whh + (size_t)(row0 + ml) * H + kb;       // z-rows of Whh
    const float* pRx = wxh + (size_t)(H + row0 + ml) * H + kb;   // r-rows of Wxh
    const float* pRh = whh + (size_t)(H + row0 + ml) * H + kb;   // r-rows of Whh
    const float* pE  = whr + (size_t)(row0 + ml) * H + kb;       // Whr
    const float* pF  = wxr + (size_t)(row0 + ml) * H + kb;       // Wxr
    const float* pX  = xt + kb;
    const float* pH  = h  + kb;

    v8f accZ = {}; v8f accR = {}; v8f accE = {}; v8f accF = {};

    // ---- software pipeline: preload chunk 0 ----
    v2f bx0  = *(const v2f*)pX;
    v2f bh0  = *(const v2f*)pH;
    v2f azx0 = *(const v2f*)pZx;
    v2f azh0 = *(const v2f*)pZh;
    v2f arx0 = *(const v2f*)pRx;
    v2f arh0 = *(const v2f*)pRh;
    v2f ae0  = *(const v2f*)pE;
    v2f af0  = *(const v2f*)pF;

    #pragma unroll 7
    for (int c = 0; c < 63; ++c) {
        // prefetch chunk c+1 (independent of the WMMAs below)
        pX += 4; pH += 4;
        pZx += 4; pZh += 4; pRx += 4; pRh += 4; pE += 4; pF += 4;
        v2f bx1  = *(const v2f*)pX;
        v2f bh1  = *(const v2f*)pH;
        v2f azx1 = *(const v2f*)pZx;
        v2f azh1 = *(const v2f*)pZh;
        v2f arx1 = *(const v2f*)pRx;
        v2f arh1 = *(const v2f*)pRh;
        v2f ae1  = *(const v2f*)pE;
        v2f af1  = *(const v2f*)pF;

        // compute chunk c
        accZ = wmma_f32(azx0, bx0, accZ);   // Wxh_z @ xt
        accZ = wmma_f32(azh0, bh0, accZ);   // + Whh_z @ h
        accR = wmma_f32(arx0, bx0, accR);   // Wxh_r @ xt
        accR = wmma_f32(arh0, bh0, accR);   // + Whh_r @ h
        accE = wmma_f32(ae0,  bh0, accE);   // Whr @ h
        accF = wmma_f32(af0,  bx0, accF);   // Wxr @ xt

        // rotate (elided inside the unrolled body)
        bx0 = bx1; bh0 = bh1;
        azx0 = azx1; azh0 = azh1; arx0 = arx1; arh0 = arh1;
        ae0 = ae1; af0 = af1;
    }

    // epilogue: last chunk
    accZ = wmma_f32(azx0, bx0, accZ);
    accZ = wmma_f32(azh0, bh0, accZ);
    accR = wmma_f32(arx0, bx0, accR);
    accR = wmma_f32(arh0, bh0, accR);
    accE = wmma_f32(ae0,  bh0, accE);
    accF = wmma_f32(af0,  bx0, accF);

    // D layout: VGPR g, lanes 0-15 -> row g, lanes 16-31 -> row g+8;
    // every lane in a group holds the same value (B was N-broadcast).
    if ((lane & 15) == 0) {
        const int rbase = (lane >> 4) << 3;
        #pragma unroll
        for (int g = 0; g < 8; ++g) {
            red[wave][0][rbase + g] = accZ[g];
            red[wave][1][rbase + g] = accR[g];
            red[wave][2][rbase + g] = accE[g];
            red[wave][3][rbase + g] = accF[g];
        }
    }
    __syncthreads();

    if (tid < 16) {
        const int row = row0 + tid;
        float sZ = 0.f, sR = 0.f, sE = 0.f, sF = 0.f;
        #pragma unroll
        for (int w = 0; w < 8; ++w) {
            sZ += red[w][0][tid];
            sR += red[w][1][tid];
            sE += red[w][2][tid];
            sF += red[w][3][tid];
        }
        float z  = 1.0f / (1.0f + __expf(-(sZ + bxh[row]     + bhh[row])));
        float r  = 1.0f / (1.0f + __expf(-(sR + bxh[H + row] + bhh[H + row])));
        float g  = tanhf((sE + bhr[row]) * r + (sF + bxr[row]));
        float hv = h[row];
        out[row] = z * hv + (1.0f - z) * g;
    }
}

extern "C" void kernel_launch(void* const* d_in, const int* in_sizes, int n_in,
                              void* d_out, int out_size, void* d_ws, size_t ws_size,
                              hipStream_t stream) {
    (void)in_sizes; (void)n_in; (void)out_size; (void)ws_size;

    const float* x   = (const float*)d_in[0];
    const float* wxh = (const float*)d_in[1] + (size_t)LAYER * 2 * H * H;
    const float* bxh = (const float*)d_in[2] + (size_t)LAYER * 2 * H;
    const float* whh = (const float*)d_in[3] + (size_t)LAYER * 2 * H * H;
    const float* bhh = (const float*)d_in[4] + (size_t)LAYER * 2 * H;
    const float* wxr = (const float*)d_in[5] + (size_t)LAYER * H * H;
    const float* bxr = (const float*)d_in[6] + (size_t)LAYER * H;
    const float* whr = (const float*)d_in[7] + (size_t)LAYER * H * H;
    const float* bhr = (const float*)d_in[8] + (size_t)LAYER * H;

    float* out   = (float*)d_out;   // [T, H]
    float* hzero = (float*)d_ws;    // [H] zeros

    rnn_init_kernel<<<H / 256, 256, 0, stream>>>(x, out, hzero);

    for (int t = 0; t < T - 1; ++t) {
        const float* xv = out + (size_t)t * H;
        const float* hv = (t == 0) ? hzero : out + (size_t)t * H;
        float*       ot = out + (size_t)(t + 1) * H;
        rnn_step_kernel<<<H / 16, 256, 0, stream>>>(
            wxh, whh, wxr, whr, bxh, bhh, bxr, bhr, xv, hv, ot);
    }
}